// TimesBlock_8538394985140
// MI455X (gfx1250) — compile-verified
//
#include <hip/hip_runtime.h>
#include <math.h>

typedef float v2f __attribute__((ext_vector_type(2)));
typedef float v8f __attribute__((ext_vector_type(8)));

#define BB   4
#define TT   48
#define WDIM 8
#define HDIM 8
#define CMOD 32   // d_model
#define CFF  64   // d_ff
#define NPOS (BB*TT*WDIM*HDIM)   // 12288
#define NTAP 81

// ---------------------------------------------------------------------------
// Weight repack to [tap][parity][j=K/4][o][2]:  element = scale * w3[o][K][taps]
// with the 1x1x1x1 conv folded into the center tap (tap==40). K is the GEMM
// contraction index == input channel. parity selects the K-pair held by lane
// halves (lanes 0-15: K=4j,4j+1 ; lanes 16-31: K=4j+2,4j+3), so a B fragment
// is one b64 load per lane.
// ---------------------------------------------------------------------------
template<int CIN, int COUT>
__global__ void prep_weights(const float* __restrict__ w1, const float* __restrict__ w3,
                             float* __restrict__ wc, float scale) {
    const int total = NTAP * CIN * COUT;
    int e = blockIdx.x * blockDim.x + threadIdx.x;
    if (e >= total) return;
    int p   = e & 1;
    int r1  = e >> 1;
    int o   = r1 % COUT;
    int r2  = r1 / COUT;
    int j   = r2 % (CIN / 4);
    int r3  = r2 / (CIN / 4);
    int par = r3 & 1;
    int tap = r3 >> 1;
    int K   = 4 * j + 2 * par + p;          // input channel
    int dh = tap % 3, dw = (tap / 3) % 3, dp = (tap / 9) % 3, dl = tap / 27;
    float v = scale * w3[((((o * CIN + K) * 3 + dl) * 3 + dp) * 3 + dw) * 3 + dh];
    if (tap == 40) v += scale * w1[o * CIN + K];
    wc[e] = v;
}

__global__ void copy_init(const float* __restrict__ src, float* __restrict__ dst, int n) {
    int i = blockIdx.x * blockDim.x + threadIdx.x;
    if (i < n) dst[i] = src[i];
}

__global__ void zero_buf(float* __restrict__ dst, int n) {
    int i = blockIdx.x * blockDim.x + threadIdx.x;
    if (i < n) dst[i] = 0.f;
}

// ---------------------------------------------------------------------------
// Stage x into a halo-padded (B, L+2, P+2, 10, 10) buffer with channels stored
// in K-pair-parity-split order: slot = ((K>>1)&1)*(C/2) + (K>>2)*2 + (K&1).
// Each lane's per-tap A data is then a contiguous C/2-float run -> no boundary
// checks and plain b64 loads in the conv kernel.
// ---------------------------------------------------------------------------
template<int PERIOD>
__global__ void pad_x_kernel(const float* __restrict__ x, float* __restrict__ xp) {
    constexpr int L = TT / PERIOD, LP = L + 2, PP = PERIOD + 2;
    const int n = BB * LP * PP * 100 * CMOD;
    int e = blockIdx.x * blockDim.x + threadIdx.x;
    if (e >= n) return;
    int slot = e & 31;
    int q    = e >> 5;
    int hh = q % 10;
    int ww = (q / 10) % 10;
    int pc = (q / 100) % PP;
    int lc = (q / (100 * PP)) % LP;
    int b  =  q / (100 * PP * LP);
    float v = 0.f;
    if (lc >= 1 && lc <= L && pc >= 1 && pc <= PERIOD &&
        ww >= 1 && ww <= 8 && hh >= 1 && hh <= 8) {
        int half = slot >> 4, rr = slot & 15;
        int K = (rr >> 1) * 4 + half * 2 + (rr & 1);
        int t = (lc - 1) * PERIOD + (pc - 1);
        v = x[(((b * TT + t) * 8 + (ww - 1)) * 8 + (hh - 1)) * CMOD + K];
    }
    xp[e] = v;
}

// ---------------------------------------------------------------------------
// Implicit-GEMM 4D conv on the padded, channel-split buffer with f32 WMMA.
// One wave = 16 positions (fixed b,t; w in {w0,w0+1}, h in 0..7) x NT*16 oc.
// PERIOD is a template arg -> padded strides/tap offsets are compile-time;
// no validity checks anywhere (halo absorbs edges).
// ---------------------------------------------------------------------------
template<int CIN, int COUT, int NT, int PERIOD, bool GELU_OUT, bool ACCUM_OUT>
__global__ __launch_bounds__(128)
void conv4d_wmma(const float* __restrict__ xin, const float* __restrict__ wcw,
                 float* __restrict__ outp) {
    constexpr int L = TT / PERIOD, LP = L + 2, PP = PERIOD + 2;
    const int lane  = threadIdx.x & 31;
    const int tile  = blockIdx.x * (blockDim.x >> 5) + (threadIdx.x >> 5); // 0..767
    const int wt    = tile & 3;
    const int bt    = tile >> 2;
    const int t     = bt % TT;
    const int b     = bt / TT;
    const int w0    = wt * 2;
    const int l     = t / PERIOD;
    const int pq    = t % PERIOD;

    const int m     = lane & 15;
    const int hiK   = lane >> 4;      // parity: which K pair this lane half holds
    const int nlane = lane & 15;

    // padded-position of tile origin (w=w0+1, h=1) and of this lane's A row
    const int qtile = (((b * LP + (l + 1)) * PP + (pq + 1)) * 10 + (w0 + 1)) * 10 + 1;
    const int qlane = qtile + (m >> 3) * 10 + (m & 7);
    const float* __restrict__ xw = xin + (long)qlane * CIN + hiK * (CIN / 2);

    v8f acc[NT];
    #pragma unroll
    for (int nt = 0; nt < NT; ++nt) acc[nt] = (v8f){0.f,0.f,0.f,0.f,0.f,0.f,0.f,0.f};

    #pragma unroll 1
    for (int il = 0; il < 3; ++il) {
        #pragma unroll 1
        for (int ip = 0; ip < 3; ++ip) {
            #pragma unroll 1
            for (int iw = 0; iw < 3; ++iw) {
                #pragma unroll
                for (int ih = 0; ih < 3; ++ih) {
                    const int tap = ((il * 3 + ip) * 3 + iw) * 3 + ih;
                    const int dq  = ((il - 1) * PP + (ip - 1)) * 100 + (iw - 1) * 10 + (ih - 1);
                    const float* __restrict__ xp = xw + (long)dq * CIN;
                    const float* __restrict__ wp =
                        wcw + ((long)tap * 2 + hiK) * (CIN / 4) * COUT * 2 + nlane * 2;
                    #pragma unroll
                    for (int j = 0; j < CIN / 4; ++j) {
                        const v2f a = *(const v2f*)(xp + 2 * j);   // contiguous K run
                        #pragma unroll
                        for (int nt = 0; nt < NT; ++nt) {
                            const v2f bf = *(const v2f*)(wp + (j * COUT + nt * 16) * 2);
                            acc[nt] = __builtin_amdgcn_wmma_f32_16x16x4_f32(
                                false, a, false, bf, (short)0, acc[nt], false, false);
                        }
                    }
                }
            }
        }
    }

    if (GELU_OUT) {
        // write gelu(acc) into the padded h buffer (channel-split layout for stage 2)
        #pragma unroll
        for (int nt = 0; nt < NT; ++nt) {
            const int o    = nt * 16 + nlane;
            const int slot = ((o >> 1) & 1) * (COUT / 2) + (o >> 2) * 2 + (o & 1);
            #pragma unroll
            for (int r = 0; r < 8; ++r) {
                const int  mr = r + hiK * 8;
                const long q  = qtile + (mr >> 3) * 10 + (mr & 7);
                float v = acc[nt][r];
                v = 0.5f * v * (1.0f + erff(v * 0.70710678118654752f));
                outp[q * COUT + slot] = v;
            }
        }
    } else {
        // accumulate into the (B,T,W,H,C) output; unique (pos,oc) per wave
        #pragma unroll
        for (int nt = 0; nt < NT; ++nt) {
            const int o = nt * 16 + nlane;
            #pragma unroll
            for (int r = 0; r < 8; ++r) {
                const int  mr = r + hiK * 8;
                const long s  = ((long)(b * TT + t) * 8 + w0 + (mr >> 3)) * 8 + (mr & 7);
                if (ACCUM_OUT) outp[s * COUT + o] += acc[nt][r];
                else           outp[s * COUT + o]  = acc[nt][r];
            }
        }
    }
}

// ---------------------------------------------------------------------------
template<int PERIOD>
static void run_period(const float* xpad_src_x, const float* wc1, const float* wc2,
                       float* xpad, float* hpad, float* out, hipStream_t stream) {
    constexpr int L = TT / PERIOD, LP = L + 2, PP = PERIOD + 2;
    constexpr int NPAD = BB * LP * PP * 100;          // padded positions
    pad_x_kernel<PERIOD><<<(NPAD * CMOD + 255) / 256, 256, 0, stream>>>(xpad_src_x, xpad);
    zero_buf<<<(NPAD * CFF + 255) / 256, 256, 0, stream>>>(hpad, NPAD * CFF);
    // 768 waves each: 192 blocks x 4 waves
    conv4d_wmma<CMOD, CFF, 4, PERIOD, true,  false><<<192, 128, 0, stream>>>(xpad, wc1, hpad);
    conv4d_wmma<CFF, CMOD, 2, PERIOD, false, true ><<<192, 128, 0, stream>>>(hpad, wc2, out);
}

extern "C" void kernel_launch(void* const* d_in, const int* in_sizes, int n_in,
                              void* d_out, int out_size, void* d_ws, size_t ws_size,
                              hipStream_t stream) {
    const float* x   = (const float*)d_in[0];
    const float* w10 = (const float*)d_in[1];   // (64,32,1,1,1,1)
    const float* w11 = (const float*)d_in[2];   // (64,32,3,3,3,3)
    const float* w20 = (const float*)d_in[3];   // (32,64,1,1,1,1)
    const float* w21 = (const float*)d_in[4];   // (32,64,3,3,3,3)
    float* out = (float*)d_out;

    const int WC1 = NTAP * CMOD * CFF;   // 165888 floats
    const int WC2 = NTAP * CFF * CMOD;   // 165888 floats
    const int XPAD_MAX = BB * 26 * 4 * 100 * CMOD;   // period=2 is largest: 1,331,200
    float* wc1  = (float*)d_ws;
    float* wc2  = wc1 + WC1;
    float* xpad = wc2 + WC2;
    float* hpad = xpad + XPAD_MAX;       // up to 2,662,400 floats

    // 0.5 folds the inception /2; stage-2 additionally folds the 1/3 softmax weight
    prep_weights<CMOD, CFF><<<(WC1 + 255) / 256, 256, 0, stream>>>(w10, w11, wc1, 0.5f);
    prep_weights<CFF, CMOD><<<(WC2 + 255) / 256, 256, 0, stream>>>(w20, w21, wc2, 1.0f / 6.0f);
    copy_init<<<(NPOS * CMOD + 255) / 256, 256, 0, stream>>>(x, out, NPOS * CMOD); // residual

    run_period<2>(x, wc1, wc2, xpad, hpad, out, stream);
    run_period<3>(x, wc1, wc2, xpad, hpad, out, stream);
    run_period<4>(x, wc1, wc2, xpad, hpad, out, stream);
}